// SAMLIF_44143673868833
// MI455X (gfx1250) — compile-verified
//
#include <hip/hip_runtime.h>

#define T_STEPS 5
#define BS      16
#define CCH     128
#define HW      4096          // 64*64
#define NROWS   80            // T_STEPS * BS

typedef float v2f __attribute__((ext_vector_type(2)));
typedef float v8f __attribute__((ext_vector_type(8)));

// ---------------------------------------------------------------------------
// Kernel 1: ga[n*C+c] = mean(|x[n,c,:,:]|)   (one block per (n,c), 10240 blocks)
// Streams 168 MB with b128 loads; lines stay RT so they remain L2-resident
// for kernel 3's re-read (168 MB < 192 MB L2).
// ---------------------------------------------------------------------------
__global__ __launch_bounds__(256) void k_ga(const float* __restrict__ x,
                                            float* __restrict__ ga) {
    __shared__ float red[256];
    const int blk = blockIdx.x;                       // n*C + c
    const int tid = threadIdx.x;
    const float4* xp = (const float4*)(x + (size_t)blk * HW);   // 1024 float4 per block
    float s = 0.f;
#pragma unroll
    for (int i = 0; i < 4; ++i) {
        float4 v = xp[tid + 256 * i];
        s += fabsf(v.x) + fabsf(v.y) + fabsf(v.z) + fabsf(v.w);
    }
    red[tid] = s;
    __syncthreads();
#pragma unroll
    for (int off = 128; off > 0; off >>= 1) {
        if (tid < off) red[tid] += red[tid + off];
        __syncthreads();
    }
    if (tid == 0) ga[blk] = red[0] * (1.0f / HW);
}

// ---------------------------------------------------------------------------
// Kernel 2: the tiny MLP + threshold mask on the matrix pipe, full f32 via
// V_WMMA_F32_16X16X4_F32. One block, 5 waves; wave w owns rows 16w..16w+15.
//   GEMM1: h_pre = ga(80x128) @ w1^T(128x4)  -> 32 chained K=4 WMMAs/wave
//   GEMM2: m_pre = h(80x4)    @ w2^T(4x128)  -> 8 N-tiles, 1 WMMA each
//   mask  = (sigmoid(m_pre + b2) > 0.5) == (m_pre + b2 > 0)
// B operand of GEMM1 is staged zero-padded in LDS so the inner loop is a
// branch-free ds_load_b64 (avoids exec-mask diamonds around scalar loads).
// ---------------------------------------------------------------------------
__global__ __launch_bounds__(160) void k_mlp(const float* __restrict__ ga,
                                             const float* __restrict__ w1,
                                             const float* __restrict__ b1,
                                             const float* __restrict__ w2,
                                             const float* __restrict__ b2,
                                             float* __restrict__ mask) {
    __shared__ float w1p[16 * CCH];                   // B padded: rows 4..15 = 0 (8 KB)
    __shared__ float hbuf[NROWS * 4];                 // 80 x 4 hidden activations
    const int tid  = threadIdx.x;
    const int wave = tid >> 5;                        // 0..4 -> M tile
    const int lane = tid & 31;
    const int half = lane >> 4;                       // selects K pair {0,1} vs {2,3}
    const int l15  = lane & 15;                       // N index / M-in-tile
    const int row  = wave * 16 + l15;                 // my A-matrix row

    // cooperative fill of padded B (16x128): w1p[n*CCH+k] = (n<4) ? w1[n][k] : 0
    for (int i = tid; i < 16 * CCH; i += 160) {
        const int n = i >> 7;                         // i / CCH
        w1p[i] = (n < 4) ? w1[i] : 0.f;               // n<4 => i == n*CCH+k
    }
    __syncthreads();

    // ---- GEMM1: 16x16 tile (only N<4 valid), K=128 in steps of 4 ----
    v8f acc = {};
#pragma unroll
    for (int kk = 0; kk < 32; ++kk) {
        const int k0 = kk * 4 + half * 2;
        v2f a, b;
        // A (16x4 f32): VGPR0 = K = 2*half, VGPR1 = K = 2*half+1, M = lane&15
        a.x = ga[row * CCH + k0];
        a.y = ga[row * CCH + k0 + 1];
        // B (4x16): B[k][n] = w1p[n][k], zero-padded for n>=4 (branch-free LDS read)
        b.x = w1p[l15 * CCH + k0];
        b.y = w1p[l15 * CCH + k0 + 1];
        acc = __builtin_amdgcn_wmma_f32_16x16x4_f32(
            /*neg_a=*/false, a, /*neg_b=*/false, b,
            /*c_mod=*/(short)0, acc, /*reuse_a=*/false, /*reuse_b=*/false);
    }
    // C/D layout: VGPR r, lane-half -> (M = 16w + r + 8*half, N = l15)
#pragma unroll
    for (int r = 0; r < 8; ++r) {
        if (l15 < 4) {
            const int m = wave * 16 + r + 8 * half;
            const float h = acc[r] + b1[l15];
            hbuf[m * 4 + l15] = (h > 0.f) ? h : 0.f;  // relu
        }
    }
    __syncthreads();

    // ---- GEMM2: 8 N-tiles of 16, K=4 (single WMMA per tile) ----
#pragma unroll
    for (int nt = 0; nt < 8; ++nt) {
        v2f a, b;
        a.x = hbuf[row * 4 + half * 2];
        a.y = hbuf[row * 4 + half * 2 + 1];
        const int n = nt * 16 + l15;                  // global output channel
        // B[k][n] = w2[n][k]; w2 is (128,4) row-major -> contiguous b64 load
        b.x = w2[n * 4 + half * 2];
        b.y = w2[n * 4 + half * 2 + 1];
        v8f acc2 = {};
        acc2 = __builtin_amdgcn_wmma_f32_16x16x4_f32(
            false, a, false, b, (short)0, acc2, false, false);
#pragma unroll
        for (int r = 0; r < 8; ++r) {
            const int m = wave * 16 + r + 8 * half;
            const float val = acc2[r] + b2[n];
            mask[m * CCH + n] = (val > 0.f) ? 1.f : 0.f;   // sigmoid(v)>0.5 <=> v>0
        }
    }
}

// ---------------------------------------------------------------------------
// Kernel 3: LIF scan. u carried in registers across the 5 timesteps; x read
// once (mostly L2 hits after kernel 1), out written once. float4 b128 traffic.
// Block handles one quarter of one (b,c) plane: 2048 * 4 = 8192 blocks.
// ---------------------------------------------------------------------------
__global__ __launch_bounds__(256) void k_lif(const float* __restrict__ x,
                                             const float* __restrict__ mask,
                                             float* __restrict__ out) {
    const int blk   = blockIdx.x;
    const int bc    = blk >> 2;                       // (b,c) pair, 0..2047
    const int chunk = blk & 3;
    const int b     = bc >> 7;
    const int c     = bc & 127;
    const int pix4  = chunk * 256 + threadIdx.x;      // float4 index in 0..1023

    const float4* xp = (const float4*)x;
    float4* op = (float4*)out;

    float4 u = {0.f, 0.f, 0.f, 0.f};
#pragma unroll
    for (int t = 0; t < T_STEPS; ++t) {
        const int n = t * BS + b;
        const size_t base = (size_t)(n * CCH + c) * (HW / 4) + pix4;
        const float4 xt = xp[base];
        const float m = mask[n * CCH + c];
        // u = TAU*u*(1 - spike(u_old)) + xt ; out = spike(u_new) * m
        u.x = ((u.x > 1.f) ? 0.f : 0.5f * u.x) + xt.x;
        u.y = ((u.y > 1.f) ? 0.f : 0.5f * u.y) + xt.y;
        u.z = ((u.z > 1.f) ? 0.f : 0.5f * u.z) + xt.z;
        u.w = ((u.w > 1.f) ? 0.f : 0.5f * u.w) + xt.w;
        float4 o;
        o.x = (u.x > 1.f) ? m : 0.f;
        o.y = (u.y > 1.f) ? m : 0.f;
        o.z = (u.z > 1.f) ? m : 0.f;
        o.w = (u.w > 1.f) ? m : 0.f;
        op[base] = o;
    }
}

// ---------------------------------------------------------------------------
extern "C" void kernel_launch(void* const* d_in, const int* in_sizes, int n_in,
                              void* d_out, int out_size, void* d_ws, size_t ws_size,
                              hipStream_t stream) {
    const float* x  = (const float*)d_in[0];   // (80,128,64,64)
    const float* w1 = (const float*)d_in[1];   // (4,128)
    const float* b1 = (const float*)d_in[2];   // (4,)
    const float* w2 = (const float*)d_in[3];   // (128,4)
    const float* b2 = (const float*)d_in[4];   // (128,)
    float* out = (float*)d_out;

    float* ga   = (float*)d_ws;                // 80*128 f32
    float* mask = ga + NROWS * CCH;            // 80*128 f32

    k_ga <<<NROWS * CCH, 256, 0, stream>>>(x, ga);
    k_mlp<<<1, 160, 0, stream>>>(ga, w1, b1, w2, b2, mask);
    k_lif<<<2048 * 4, 256, 0, stream>>>(x, mask, out);
}